// Encoder_71777493450829
// MI455X (gfx1250) — compile-verified
//
#include <hip/hip_runtime.h>
#include <hip/hip_bf16.h>
#include <climits>

typedef float v2f __attribute__((ext_vector_type(2)));
typedef float v8f __attribute__((ext_vector_type(8)));

// ---------------------------------------------------------------------------
// Generic fp32 GEMM via V_WMMA_F32_16X16X4_F32.
// Each wave owns one 16x16 output tile; K advanced 4 per WMMA.
// A: [M,K] row-major, B: [K,N] row-major, C: [M,N] row-major.
// Requires M%16==0, N%16==0, K%4==0 (true for all calls here).
// ---------------------------------------------------------------------------
__global__ void gemm_wmma_f32(const float* __restrict__ A,
                              const float* __restrict__ B,
                              float* __restrict__ C,
                              int M, int N, int K) {
    const int lane = threadIdx.x & 31;
    const int wave = threadIdx.x >> 5;
    const int wavesPerBlock = blockDim.x >> 5;

    const int col0 = (blockIdx.x * wavesPerBlock + wave) * 16;   // wave-uniform
    const int row0 = blockIdx.y * 16;                            // wave-uniform
    if (col0 >= N || row0 >= M) return;  // uniform per wave -> EXEC stays all-1s

    const int m  = lane & 15;   // row within tile (A), col within tile (B/C/D)
    const int kh = lane >> 4;   // selects K pair {0,1} vs {2,3}

    v8f acc = {};
    const float* Arow = A + (size_t)(row0 + m) * K + 2 * kh;
    const float* Bcol = B + (size_t)(2 * kh) * N + col0 + m;

    for (int k0 = 0; k0 < K; k0 += 4) {
        v2f a = *(const v2f*)(Arow + k0);          // A[row0+m][k0+2kh : +1]
        v2f b;
        b[0] = Bcol[(size_t)k0 * N];               // B[k0+2kh  ][col0+m]
        b[1] = Bcol[(size_t)k0 * N + N];           // B[k0+2kh+1][col0+m]
        acc = __builtin_amdgcn_wmma_f32_16x16x4_f32(
            /*neg_a=*/false, a, /*neg_b=*/false, b,
            /*c_mod=*/(short)0, acc, /*reuse_a=*/false, /*reuse_b=*/false);
    }

    // D layout: VGPR r -> row (row0 + r + 8*kh), col (col0 + m)
    float* Crow = C + (size_t)(row0 + 8 * kh) * N + col0 + m;
#pragma unroll
    for (int r = 0; r < 8; ++r)
        Crow[(size_t)r * N] = acc[r];
}

// ---------------------------------------------------------------------------
// Attention logits: aL[n,h] = sum_f xl[n,h,f]*attL[h,f]; same for aR.
// xl flat [N,H,F]; one thread per (n,h).
// ---------------------------------------------------------------------------
__global__ void k_att(const float* __restrict__ xl,
                      const float* __restrict__ attL,
                      const float* __restrict__ attR,
                      float* __restrict__ aL, float* __restrict__ aR,
                      int N, int F) {
    int idx = blockIdx.x * blockDim.x + threadIdx.x;   // n*8 + h
    if (idx >= N * 8) return;
    int h = idx & 7;
    const float* row = xl + (size_t)idx * F;
    const float* al = attL + h * F;
    const float* ar = attR + h * F;
    float sl = 0.f, sr = 0.f;
    for (int f = 0; f < F; ++f) {
        float v = row[f];
        sl += v * al[f];
        sr += v * ar[f];
    }
    aL[idx] = sl;
    aR[idx] = sr;
}

// Order-preserving float<->int encoding for atomic max on signed ints.
__device__ __forceinline__ int f2ord(float f) {
    int b = __float_as_int(f);
    return b >= 0 ? b : (b ^ 0x7fffffff);
}
__device__ __forceinline__ float ord2f(int b) {
    return __int_as_float(b >= 0 ? b : (b ^ 0x7fffffff));
}

// Pass A: e = leaky_relu(aL[src]+aR[dst], 0.2); atomic segment max over dst.
__global__ void k_edge_score(const float* __restrict__ aL,
                             const float* __restrict__ aR,
                             const long long* __restrict__ src,
                             const long long* __restrict__ dst,
                             float* __restrict__ e, int* __restrict__ mbits,
                             long long E) {
    long long idx = blockIdx.x * (long long)blockDim.x + threadIdx.x; // edge*8+h
    if (idx >= E * 8) return;
    long long ed = idx >> 3;
    int h = (int)(idx & 7);
    long long s = src[ed], d = dst[ed];
    float v = aL[s * 8 + h] + aR[d * 8 + h];
    v = v > 0.f ? v : 0.2f * v;
    e[idx] = v;
    atomicMax(mbits + d * 8 + h, f2ord(v));
}

// Pass B: e = exp(e - max[dst]); atomic segment sum of e over dst.
__global__ void k_edge_exp(const long long* __restrict__ dst,
                           float* __restrict__ e,
                           const int* __restrict__ mbits,
                           float* __restrict__ denom, long long E) {
    long long idx = blockIdx.x * (long long)blockDim.x + threadIdx.x;
    if (idx >= E * 8) return;
    long long ed = idx >> 3;
    int h = (int)(idx & 7);
    long long d = dst[ed];
    float mv = ord2f(mbits[d * 8 + h]);
    float ex = __expf(e[idx] - mv);
    e[idx] = ex;
    atomicAdd(denom + d * 8 + h, ex);
}

// Pass C: agg[dst,h,f] += (e/denom[dst]) * xl[src,h,f]. One thread per (e,h,f).
__global__ void k_edge_agg(const long long* __restrict__ src,
                           const long long* __restrict__ dst,
                           const float* __restrict__ e,
                           const float* __restrict__ denom,
                           const float* __restrict__ xl,
                           float* __restrict__ agg,
                           long long E, int F) {
    long long idx = blockIdx.x * (long long)blockDim.x + threadIdx.x;
    long long total = E * 8 * F;
    if (idx >= total) return;
    long long ed = idx / (8 * F);
    int r = (int)(idx - ed * (8 * F));   // h*F + f
    int h = r / F;
    long long s = src[ed], d = dst[ed];
    float alpha = e[ed * 8 + h] / (denom[d * 8 + h] + 1e-16f);
    atomicAdd(agg + d * (long long)(8 * F) + r, alpha * xl[s * (long long)(8 * F) + r]);
}

__global__ void k_bias_relu(float* __restrict__ a, const float* __restrict__ b,
                            long long n, int C) {
    long long idx = blockIdx.x * (long long)blockDim.x + threadIdx.x;
    if (idx >= n) return;
    int c = (int)(idx % C);
    float v = a[idx] + b[c];
    a[idx] = v > 0.f ? v : 0.f;
}

// out[n,f] = mean_h agg2[n,h,f] + b2[f]
__global__ void k_final(const float* __restrict__ agg2,
                        const float* __restrict__ b2,
                        float* __restrict__ out, int N) {
    long long idx = blockIdx.x * (long long)blockDim.x + threadIdx.x;
    if (idx >= (long long)N * 64) return;
    int f = (int)(idx & 63);
    long long n = idx >> 6;
    float s = 0.f;
#pragma unroll
    for (int h = 0; h < 8; ++h)
        s += agg2[n * 512 + h * 64 + f];
    out[idx] = s * 0.125f + b2[f];
}

__global__ void k_fill_f(float* p, float v, long long n) {
    long long idx = blockIdx.x * (long long)blockDim.x + threadIdx.x;
    if (idx < n) p[idx] = v;
}
__global__ void k_fill_i(int* p, int v, long long n) {
    long long idx = blockIdx.x * (long long)blockDim.x + threadIdx.x;
    if (idx < n) p[idx] = v;
}

// ---------------------------------------------------------------------------
extern "C" void kernel_launch(void* const* d_in, const int* in_sizes, int n_in,
                              void* d_out, int out_size, void* d_ws, size_t ws_size,
                              hipStream_t stream) {
    const float* x      = (const float*)d_in[0];
    const long long* ei = (const long long*)d_in[1];
    const float* W1     = (const float*)d_in[2];
    const float* attL1  = (const float*)d_in[3];
    const float* attR1  = (const float*)d_in[4];
    const float* b1     = (const float*)d_in[5];
    const float* W2     = (const float*)d_in[6];
    const float* attL2  = (const float*)d_in[7];
    const float* attR2  = (const float*)d_in[8];
    const float* b2     = (const float*)d_in[9];
    float* out = (float*)d_out;

    const int IN = 128, H = 8, F1 = 8, F2 = 64;
    const int C1 = H * F1;   // 64
    const int C2 = H * F2;   // 512
    const int N = in_sizes[0] / IN;          // 50000
    const long long E = in_sizes[1] / 2;     // 800000
    const long long* src = ei;
    const long long* dst = ei + E;

    // workspace layout (floats)
    float* ws   = (float*)d_ws;
    float* xl1  = ws;                                  // N*64
    float* agg1 = xl1  + (size_t)N * C1;               // N*64  (becomes h)
    float* xl2  = agg1 + (size_t)N * C1;               // N*512
    float* agg2 = xl2  + (size_t)N * C2;               // N*512
    float* aL   = agg2 + (size_t)N * C2;               // N*8
    float* aR   = aL   + (size_t)N * H;                // N*8
    float* den  = aR   + (size_t)N * H;                // N*8
    int*   mb   = (int*)(den + (size_t)N * H);         // N*8 ints
    float* ebuf = (float*)(mb + (size_t)N * H);        // E*8

    const int T = 256;
    auto blocks = [&](long long n) { return (unsigned)((n + T - 1) / T); };

    // ---------------- Layer 1 ----------------
    {   // GEMM1: [N,128]x[128,64] -> xl1, 4 waves/block -> whole N-dim per block
        dim3 blk(128), grd(1, (N + 15) / 16);
        gemm_wmma_f32<<<grd, blk, 0, stream>>>(x, W1, xl1, N, C1, IN);
    }
    k_att<<<blocks((long long)N * H), T, 0, stream>>>(xl1, attL1, attR1, aL, aR, N, F1);

    k_fill_i<<<blocks((long long)N * H), T, 0, stream>>>(mb, INT_MIN, (long long)N * H);
    k_fill_f<<<blocks((long long)N * H), T, 0, stream>>>(den, 0.f, (long long)N * H);
    k_fill_f<<<blocks((long long)N * C1), T, 0, stream>>>(agg1, 0.f, (long long)N * C1);

    k_edge_score<<<blocks(E * H), T, 0, stream>>>(aL, aR, src, dst, ebuf, mb, E);
    k_edge_exp  <<<blocks(E * H), T, 0, stream>>>(dst, ebuf, mb, den, E);
    k_edge_agg  <<<blocks(E * H * F1), T, 0, stream>>>(src, dst, ebuf, den, xl1, agg1, E, F1);

    k_bias_relu<<<blocks((long long)N * C1), T, 0, stream>>>(agg1, b1, (long long)N * C1, C1);

    // ---------------- Layer 2 ----------------
    {   // GEMM2: [N,64]x[64,512] -> xl2, 8 waves/block, 4 column-blocks
        dim3 blk(256), grd((C2 + 16 * 8 - 1) / (16 * 8), (N + 15) / 16);
        gemm_wmma_f32<<<grd, blk, 0, stream>>>(agg1, W2, xl2, N, C2, C1);
    }
    k_att<<<blocks((long long)N * H), T, 0, stream>>>(xl2, attL2, attR2, aL, aR, N, F2);

    k_fill_i<<<blocks((long long)N * H), T, 0, stream>>>(mb, INT_MIN, (long long)N * H);
    k_fill_f<<<blocks((long long)N * H), T, 0, stream>>>(den, 0.f, (long long)N * H);
    k_fill_f<<<blocks((long long)N * C2), T, 0, stream>>>(agg2, 0.f, (long long)N * C2);

    k_edge_score<<<blocks(E * H), T, 0, stream>>>(aL, aR, src, dst, ebuf, mb, E);
    k_edge_exp  <<<blocks(E * H), T, 0, stream>>>(dst, ebuf, mb, den, E);
    k_edge_agg  <<<blocks(E * H * F2), T, 0, stream>>>(src, dst, ebuf, den, xl2, agg2, E, F2);

    k_final<<<blocks((long long)N * F2), T, 0, stream>>>(agg2, b2, out, N);
}